// GCN_13907104104963
// MI455X (gfx1250) — compile-verified
//
#include <hip/hip_runtime.h>
#include <hip/hip_bf16.h>

// ---------------------------------------------------------------------------
// 2-layer GCN for MI455X (gfx1250, wave32).
//   x:[N,128] f32, edge_index:[2,E] i64, W0:[128,128], b0:[128],
//   W1:[128,64], b1:[64]  ->  out:[N,64] f32 (sigmoid)
// GEMMs use V_WMMA_F32_16X16X4_F32 (full f32 precision, matches reference).
// Edge aggregation: wave-per-edge gather + f32 HW atomics (L2-resident data).
// ---------------------------------------------------------------------------

typedef float v2f __attribute__((ext_vector_type(2)));
typedef float v8f __attribute__((ext_vector_type(8)));

#define N_FIN 128
#define N_HID 128
#define N_OUT 64
#define GEMM_K 128

// ---------------- degree / normalization ----------------------------------

__global__ void gcn_init_deg(float* deg, int n) {
    int i = blockIdx.x * blockDim.x + threadIdx.x;
    if (i < n) deg[i] = 1.0f;              // self-loop contributes 1
}

__global__ void gcn_deg_accum(const long long* __restrict__ dst, float* deg, long long E) {
    long long e = (long long)blockIdx.x * blockDim.x + threadIdx.x;
    if (e < E) unsafeAtomicAdd(&deg[(int)dst[e]], 1.0f);
}

__global__ void gcn_dinv(float* deg, int n) {
    int i = blockIdx.x * blockDim.x + threadIdx.x;
    if (i < n) deg[i] = rsqrtf(deg[i]);    // deg >= 1 always
}

// ---------------- WMMA f32 GEMM: C[M,NCOLS] = A[M,128] @ B[128,NCOLS] ------
// One block = one 16-row tile of A (staged in LDS), one wave per 16-col tile.
// Requires M % 16 == 0 (true here: 50000 = 3125*16).

template <int NCOLS>
__global__ __launch_bounds__(NCOLS * 2) void gcn_wmma_gemm(
    const float* __restrict__ A, const float* __restrict__ B,
    float* __restrict__ C, int M)
{
    constexpr int K = GEMM_K;
    constexpr int LDA = K + 4;                      // pad: conflict-free LDS
    __shared__ float sA[16 * LDA];

    const int row0 = blockIdx.x * 16;
    const int nthreads = NCOLS * 2;

    // cooperative load of the 16xK A tile
    for (int i = threadIdx.x; i < 16 * K; i += nthreads) {
        const int r = i >> 7;                        // i / 128
        const int c = i & (K - 1);
        sA[r * LDA + c] = A[(size_t)(row0 + r) * K + c];
    }
    __syncthreads();

    const int wave = threadIdx.x >> 5;               // 0 .. NCOLS/16-1
    const int lane = threadIdx.x & 31;
    const int n0   = wave * 16;
    const int m    = lane & 15;                      // A row within tile
    const int kh   = (lane >> 4) << 1;               // 0 (lanes 0-15) or 2

    v8f acc = {};
#pragma unroll
    for (int k0 = 0; k0 < K; k0 += 4) {
        v2f a, b;
        // A 16x4 f32 layout: VGPR0 = K {0|2}, VGPR1 = K {1|3} per half-wave
        a.x = sA[m * LDA + k0 + kh];
        a.y = sA[m * LDA + k0 + kh + 1];
        // B 4x16 f32 layout (row striped across lanes, mirrored halves)
        b.x = B[(size_t)(k0 + kh) * NCOLS + n0 + m];
        b.y = B[(size_t)(k0 + kh + 1) * NCOLS + n0 + m];
        acc = __builtin_amdgcn_wmma_f32_16x16x4_f32(
            /*neg_a=*/false, a, /*neg_b=*/false, b,
            /*c_mod=*/(short)0, acc, /*reuse_a=*/false, /*reuse_b=*/false);
    }

    // C 16x16 f32 layout: VGPR r -> row r (lanes 0-15) / row r+8 (lanes 16-31)
    const int rowAdd = (lane < 16) ? 0 : 8;
    const int col    = n0 + (lane & 15);
#pragma unroll
    for (int r = 0; r < 8; ++r) {
        C[(size_t)(row0 + r + rowAdd) * NCOLS + col] = acc[r];
    }
}

// ---------------- self-loop init: agg[i,:] = h[i,:] * dinv[i]^2 ------------

template <int F>
__global__ void gcn_selfloop_init(const float* __restrict__ h,
                                  const float* __restrict__ dinv,
                                  float* __restrict__ agg, int n)
{
    long long idx = (long long)blockIdx.x * blockDim.x + threadIdx.x;
    if (idx < (long long)n * F) {
        const int i = (int)(idx / F);
        const float s = dinv[i];
        agg[idx] = h[idx] * s * s;
    }
}

// ---------------- edge scatter: one wave per edge --------------------------
// agg[dst,:] += h[src,:] * dinv[src]*dinv[dst]; lane handles F/32 floats.

template <int F>
__global__ void gcn_edge_agg(const float* __restrict__ h,
                             const float* __restrict__ dinv,
                             const long long* __restrict__ src,
                             const long long* __restrict__ dst,
                             float* __restrict__ agg, long long E)
{
    constexpr int VEC = F / 32;                      // 4 (F=128) or 2 (F=64)
    const int lane = threadIdx.x & 31;
    const long long e = (long long)blockIdx.x * (blockDim.x >> 5) + (threadIdx.x >> 5);
    if (e >= E) return;

    const int s = (int)src[e];
    const int d = (int)dst[e];
    const float norm = dinv[s] * dinv[d];

    const float* hp = h   + (size_t)s * F + lane * VEC;
    float*       ap = agg + (size_t)d * F + lane * VEC;

    if constexpr (VEC == 4) {
        const float4 hv = *reinterpret_cast<const float4*>(hp);
        unsafeAtomicAdd(&ap[0], hv.x * norm);
        unsafeAtomicAdd(&ap[1], hv.y * norm);
        unsafeAtomicAdd(&ap[2], hv.z * norm);
        unsafeAtomicAdd(&ap[3], hv.w * norm);
    } else {
        const float2 hv = *reinterpret_cast<const float2*>(hp);
        unsafeAtomicAdd(&ap[0], hv.x * norm);
        unsafeAtomicAdd(&ap[1], hv.y * norm);
    }
}

// ---------------- epilogues ------------------------------------------------

template <int F>
__global__ void gcn_bias_relu(float* __restrict__ a, const float* __restrict__ b, int n)
{
    long long idx = (long long)blockIdx.x * blockDim.x + threadIdx.x;
    if (idx < (long long)n * F) {
        const float v = a[idx] + b[(int)(idx % F)];
        a[idx] = v > 0.0f ? v : 0.0f;
    }
}

template <int F>
__global__ void gcn_bias_sigmoid(float* __restrict__ a, const float* __restrict__ b, int n)
{
    long long idx = (long long)blockIdx.x * blockDim.x + threadIdx.x;
    if (idx < (long long)n * F) {
        const float v = a[idx] + b[(int)(idx % F)];
        a[idx] = 1.0f / (1.0f + expf(-v));
    }
}

// ---------------- host-side launcher ---------------------------------------

extern "C" void kernel_launch(void* const* d_in, const int* in_sizes, int n_in,
                              void* d_out, int out_size, void* d_ws, size_t ws_size,
                              hipStream_t stream)
{
    const float*     x   = (const float*)d_in[0];
    const long long* ei  = (const long long*)d_in[1];
    const float*     W0  = (const float*)d_in[2];
    const float*     b0  = (const float*)d_in[3];
    const float*     W1  = (const float*)d_in[4];
    const float*     b1  = (const float*)d_in[5];
    float*           out = (float*)d_out;

    const int       N    = in_sizes[0] / N_FIN;      // 50000 (multiple of 16)
    const long long E    = (long long)in_sizes[1] / 2;
    const long long* src = ei;
    const long long* dst = ei + E;

    // workspace layout (floats)
    float* ws   = (float*)d_ws;
    size_t off  = 0;
    float* dinv = ws + off; off += ((size_t)N + 63) & ~(size_t)63;
    float* h1   = ws + off; off += (size_t)N * N_HID;   // gemm1 out
    float* agg1 = ws + off; off += (size_t)N * N_HID;   // agg1 -> relu in place
    float* h2   = ws + off; off += (size_t)N * N_OUT;   // gemm2 out

    const int T = 256;

    // normalization coefficients
    gcn_init_deg<<<(N + T - 1) / T, T, 0, stream>>>(dinv, N);
    gcn_deg_accum<<<(int)((E + T - 1) / T), T, 0, stream>>>(dst, dinv, E);
    gcn_dinv<<<(N + T - 1) / T, T, 0, stream>>>(dinv, N);

    // ---- layer 1: h1 = x @ W0 ; agg ; relu(+b0)
    gcn_wmma_gemm<N_HID><<<N / 16, N_HID * 2, 0, stream>>>(x, W0, h1, N);
    {
        const long long tot = (long long)N * N_HID;
        gcn_selfloop_init<N_HID><<<(int)((tot + T - 1) / T), T, 0, stream>>>(h1, dinv, agg1, N);
        gcn_edge_agg<N_HID><<<(int)((E + 7) / 8), T, 0, stream>>>(h1, dinv, src, dst, agg1, E);
        gcn_bias_relu<N_HID><<<(int)((tot + T - 1) / T), T, 0, stream>>>(agg1, b0, N);
    }

    // ---- layer 2: h2 = agg1 @ W1 ; agg into d_out ; sigmoid(+b1)
    gcn_wmma_gemm<N_OUT><<<N / 16, N_OUT * 2, 0, stream>>>(agg1, W1, h2, N);
    {
        const long long tot = (long long)N * N_OUT;
        gcn_selfloop_init<N_OUT><<<(int)((tot + T - 1) / T), T, 0, stream>>>(h2, dinv, out, N);
        gcn_edge_agg<N_OUT><<<(int)((E + 7) / 8), T, 0, stream>>>(h2, dinv, src, dst, out, E);
        gcn_bias_sigmoid<N_OUT><<<(int)((tot + T - 1) / T), T, 0, stream>>>(out, b1, N);
    }
}